// LinearAttention_51642686767976
// MI455X (gfx1250) — compile-verified
//
#include <hip/hip_runtime.h>

typedef __attribute__((ext_vector_type(2))) float v2f;
typedef __attribute__((ext_vector_type(8))) float v8f;
typedef unsigned int u32x4 __attribute__((ext_vector_type(4)));
typedef int i32x4 __attribute__((ext_vector_type(4)));
typedef int i32x8 __attribute__((ext_vector_type(8)));

#define B_ 4
#define D_ 64
#define N_ 131072
#define PART_STRIDE 4160   // 4096 (KV 64x64) + 64 (K_sum)
#define LDS_S 68           // row stride (dwords): 16B-aligned rows, conflict-free frags

#if defined(__HIP_DEVICE_COMPILE__) && __has_builtin(__builtin_amdgcn_tensor_load_to_lds)
#define USE_TDM 1
#else
#define USE_TDM 0
#endif

static __device__ __forceinline__ v8f wmma4(v2f a, v2f b, v8f c) {
  // D = A(16x4) * B(4x16) + C, fp32
  return __builtin_amdgcn_wmma_f32_16x16x4_f32(false, a, false, b, (short)0, c,
                                               false, false);
}

static __device__ __forceinline__ float phi(float x) {
  // elu(x)+1 = x+1 (x>0) else exp(x)
  return x > 0.0f ? x + 1.0f : expf(x);
}

#if USE_TDM
// TDM: DMA a 64x64 f32 tile (row stride N_ elements in global) into LDS,
// padding +4 dwords after every 64 dwords so rows land at LDS stride 68.
static __device__ __forceinline__ void tdm_load_v(const float* gsrc, void* ldst) {
  unsigned long long ga = (unsigned long long)(uintptr_t)gsrc;
  unsigned int la = (unsigned int)(uintptr_t)ldst;  // LDS byte offset (addr[31:0])
  u32x4 g0;
  g0[0] = 1u;                                        // count=1, is_restore=0
  g0[1] = la;                                        // lds_addr
  g0[2] = (unsigned int)ga;                          // global_addr[31:0]
  g0[3] = (unsigned int)((ga >> 32) & 0x01FFFFFFu)   // global_addr[56:32]
        | (2u << 30);                                // type=2 ("image")
  i32x8 g1;
  g1[0] = (2 << 16)        // data_size = 4B
        | (1 << 20)        // pad_enable
        | (5 << 22)        // pad_interval: 64 dwords
        | (3 << 25);       // pad_amount: 4 dwords
  g1[1] = (64 << 16);      // tensor_dim0 = 64
  g1[2] = (64 << 16);      // tensor_dim1 = 64
  g1[3] = (64 << 16);      // tile_dim0 = 64
  g1[4] = 64;              // tile_dim1 = 64, tile_dim2 = 0
  g1[5] = N_;              // tensor_dim0_stride[31:0]
  g1[6] = 0;
  g1[7] = 0;
  i32x4 g2 = {0, 0, 0, 0};
  i32x4 g3 = {0, 0, 0, 0};
#if defined(__clang_major__) && (__clang_major__ >= 23)
  i32x8 gz = {0, 0, 0, 0, 0, 0, 0, 0};
  __builtin_amdgcn_tensor_load_to_lds(g0, g1, g2, g3, gz, 0);
#else
  __builtin_amdgcn_tensor_load_to_lds(g0, g1, g2, g3, 0);
#endif
}
#endif

// ---------------- Pass 1: KV partials + K_sum partials -------------------
__global__ __launch_bounds__(256) void la_pass1(const float* __restrict__ kin,
                                                const float* __restrict__ vin,
                                                const float* __restrict__ scaleKp,
                                                float* __restrict__ partials) {
  __shared__ __align__(16) float ldsK[D_ * LDS_S];
  __shared__ __align__(16) float ldsV[2][D_ * LDS_S];

  const int t    = threadIdx.x;
  const int lane = t & 31;
  const int wid  = t >> 5;
  const int b    = blockIdx.y;
  const float scaleK = scaleKp[0];

  // wave owns 2 of the 16 (Mt,Nt) tiles of the 64x64 KV, sharing the A-frag
  const int tid0 = wid * 2;
  const int Mt   = tid0 >> 2;
  const int Nt0  = tid0 & 3;
  const int Nt1  = Nt0 + 1;

  const int r   = lane & 15;
  const int hi2 = (lane >> 4) << 1;
  const int hi8 = (lane >> 4) << 3;

  v8f acc0 = {}, acc1 = {};
  float ksum = 0.0f;

  const size_t gbase = (size_t)b * D_ * N_;
  const int nchunks = N_ / 64;

  float4 kreg[4];
#if !USE_TDM
  float4 vreg[4];
#endif

  // ---- prologue: prefetch chunk blockIdx.x ----
  int c = blockIdx.x;
  if (c < nchunks) {
    #pragma unroll
    for (int i = 0; i < 4; ++i) {
      int idx4 = t + 256 * i;
      int d  = idx4 >> 4;
      int c4 = idx4 & 15;
      size_t g = gbase + (size_t)d * N_ + (size_t)c * 64 + c4 * 4;
      kreg[i] = *(const float4*)(kin + g);
#if !USE_TDM
      vreg[i] = *(const float4*)(vin + g);
#endif
    }
#if USE_TDM
    if (wid == 0) tdm_load_v(vin + gbase + (size_t)c * 64, &ldsV[0][0]);
#endif
  }

  int p = 0;
  for (; c < nchunks; c += gridDim.x, p ^= 1) {
    const int cn = c + (int)gridDim.x;
    const bool has_next = cn < nchunks;

#if USE_TDM
    // issue next V tile DMA into the other buffer (freed by last iter's barrier)
    if (has_next && wid == 0)
      tdm_load_v(vin + gbase + (size_t)cn * 64, &ldsV[p ^ 1][0]);
#endif

    // stage phi(k*scaleK) from regs into LDS (and v on the fallback path)
    #pragma unroll
    for (int i = 0; i < 4; ++i) {
      int idx4 = t + 256 * i;
      int d  = idx4 >> 4;
      int c4 = idx4 & 15;
      float4 pk;
      pk.x = phi(kreg[i].x * scaleK); pk.y = phi(kreg[i].y * scaleK);
      pk.z = phi(kreg[i].z * scaleK); pk.w = phi(kreg[i].w * scaleK);
      *(float4*)(&ldsK[d * LDS_S + c4 * 4]) = pk;
#if !USE_TDM
      *(float4*)(&ldsV[p][d * LDS_S + c4 * 4]) = vreg[i];
#endif
    }

    // prefetch next chunk's k (latency hidden behind this chunk's WMMAs)
    if (has_next) {
      #pragma unroll
      for (int i = 0; i < 4; ++i) {
        int idx4 = t + 256 * i;
        int d  = idx4 >> 4;
        int c4 = idx4 & 15;
        size_t g = gbase + (size_t)d * N_ + (size_t)cn * 64 + c4 * 4;
        kreg[i] = *(const float4*)(kin + g);
#if !USE_TDM
        vreg[i] = *(const float4*)(vin + g);
#endif
      }
    }

#if USE_TDM
    // wait for the *current* tile's DMA (the next one may stay in flight)
    if (has_next) __builtin_amdgcn_s_wait_tensorcnt(1);
    else          __builtin_amdgcn_s_wait_tensorcnt(0);
#endif
    __syncthreads();

    // K_sum partial: thread t<64 owns row d=t
    if (t < 64) {
      float s = 0.0f;
      #pragma unroll 8
      for (int j = 0; j < 64; ++j) s += ldsK[t * LDS_S + j];
      ksum += s;
    }

    // KV += phiK(64 x 64chunk) @ V^T  via 16 K-steps of fp32 WMMA
    #pragma unroll
    for (int kk = 0; kk < 16; ++kk) {
      int off = kk * 4 + hi2;
      v2f a  = *(const v2f*)(&ldsK[(Mt  * 16 + r) * LDS_S + off]);
      v2f b0 = *(const v2f*)(&ldsV[p][(Nt0 * 16 + r) * LDS_S + off]);
      v2f b1 = *(const v2f*)(&ldsV[p][(Nt1 * 16 + r) * LDS_S + off]);
      acc0 = wmma4(a, b0, acc0);
      acc1 = wmma4(a, b1, acc1);
    }
    __syncthreads();
  }

  // per-workgroup partial (deterministic: no float atomics)
  float* pp = partials + ((size_t)b * gridDim.x + blockIdx.x) * PART_STRIDE;
  #pragma unroll
  for (int j = 0; j < 8; ++j) {
    int d = Mt * 16 + j + hi8;
    pp[d * 64 + Nt0 * 16 + r] = acc0[j];
    pp[d * 64 + Nt1 * 16 + r] = acc1[j];
  }
  if (t < 64) pp[4096 + t] = ksum;
}

// ---------------- Fixed-order reduction of partials ----------------------
__global__ __launch_bounds__(256) void la_reduce(const float* __restrict__ partials,
                                                 float* __restrict__ red, int wgs1) {
  int i = blockIdx.x * blockDim.x + threadIdx.x;
  if (i >= B_ * PART_STRIDE) return;
  int b   = i / PART_STRIDE;
  int off = i - b * PART_STRIDE;
  const float* base = partials + (size_t)b * wgs1 * PART_STRIDE + off;
  float s = 0.0f;
  for (int w = 0; w < wgs1; ++w) s += base[(size_t)w * PART_STRIDE];
  red[i] = s;
}

// ---------------- Pass 2: V_new = phiQ @ KV, scaled by Z -----------------
__global__ __launch_bounds__(256) void la_pass2(const float* __restrict__ qin,
                                                const float* __restrict__ red,
                                                float* __restrict__ out) {
  __shared__ __align__(16) float kv[D_ * LDS_S];   // 4352 dwords
  __shared__ float ksum[64];
  __shared__ float zbuf[128];
  __shared__ __align__(16) float qt[128 * LDS_S];  // phiQ^T; reused as vnew[64*129]
  float* vnew = qt;

  const int t    = threadIdx.x;
  const int lane = t & 31;
  const int wid  = t >> 5;
  const int b    = blockIdx.y;

  const int r   = lane & 15;
  const int hi2 = (lane >> 4) << 1;
  const int hi8 = (lane >> 4) << 3;
  const int Mt  = wid;               // 8 M-tiles of n cover the 128-n chunk

  // stage KV (64x64) and K_sum once
  const float* redb = red + (size_t)b * PART_STRIDE;
  #pragma unroll
  for (int i = 0; i < 4; ++i) {
    int idx4 = t + 256 * i;
    int d  = idx4 >> 4;
    int e4 = idx4 & 15;
    float4 x = *(const float4*)(redb + idx4 * 4);
    *(float4*)(&kv[d * LDS_S + e4 * 4]) = x;
  }
  if (t < 64) ksum[t] = redb[4096 + t];
  __syncthreads();

  const size_t gbase = (size_t)b * D_ * N_;
  const int nchunks = N_ / 128;

  float4 qreg[8];
  int c = blockIdx.x;
  if (c < nchunks) {
    #pragma unroll
    for (int i = 0; i < 8; ++i) {
      int idx4 = t + 256 * i;
      int d  = idx4 >> 5;
      int n4 = idx4 & 31;
      qreg[i] = *(const float4*)(qin + gbase + (size_t)d * N_ + (size_t)c * 128 + n4 * 4);
    }
  }

  for (; c < nchunks; c += gridDim.x) {
    // stage phi(q) transposed: qt[n][d]
    #pragma unroll
    for (int i = 0; i < 8; ++i) {
      int idx4 = t + 256 * i;
      int d  = idx4 >> 5;
      int n4 = idx4 & 31;
      qt[(n4 * 4 + 0) * LDS_S + d] = phi(qreg[i].x);
      qt[(n4 * 4 + 1) * LDS_S + d] = phi(qreg[i].y);
      qt[(n4 * 4 + 2) * LDS_S + d] = phi(qreg[i].z);
      qt[(n4 * 4 + 3) * LDS_S + d] = phi(qreg[i].w);
    }

    // prefetch next chunk's q (hidden behind WMMA phase)
    const int cn = c + (int)gridDim.x;
    if (cn < nchunks) {
      #pragma unroll
      for (int i = 0; i < 8; ++i) {
        int idx4 = t + 256 * i;
        int d  = idx4 >> 5;
        int n4 = idx4 & 31;
        qreg[i] = *(const float4*)(qin + gbase + (size_t)d * N_ + (size_t)cn * 128 + n4 * 4);
      }
    }
    __syncthreads();

    // Z[n] = 1/(phiQ[n,:].K_sum + eps)
    if (t < 128) {
      float s = 0.0f;
      #pragma unroll 8
      for (int d = 0; d < 64; ++d) s += qt[t * LDS_S + d] * ksum[d];
      zbuf[t] = 1.0f / (s + 1e-8f);
    }

    // (128x64) = phiQ_chunk @ KV : per wave, 1 M-tile x 4 N-tiles, 16 K-steps
    v8f a0 = {}, a1 = {}, a2 = {}, a3 = {};
    #pragma unroll
    for (int kk = 0; kk < 16; ++kk) {
      int krow = kk * 4 + hi2;
      v2f afr = *(const v2f*)(&qt[(Mt * 16 + r) * LDS_S + krow]);
      v2f b0, b1, b2, b3;
      b0.x = kv[krow * LDS_S +  0 + r]; b0.y = kv[(krow + 1) * LDS_S +  0 + r];
      b1.x = kv[krow * LDS_S + 16 + r]; b1.y = kv[(krow + 1) * LDS_S + 16 + r];
      b2.x = kv[krow * LDS_S + 32 + r]; b2.y = kv[(krow + 1) * LDS_S + 32 + r];
      b3.x = kv[krow * LDS_S + 48 + r]; b3.y = kv[(krow + 1) * LDS_S + 48 + r];
      a0 = wmma4(afr, b0, a0);
      a1 = wmma4(afr, b1, a1);
      a2 = wmma4(afr, b2, a2);
      a3 = wmma4(afr, b3, a3);
    }
    __syncthreads();   // qt reads done -> safe to overwrite with vnew

    // transpose tiles into vnew[e][n_local] (stride 129: conflict-light)
    #pragma unroll
    for (int j = 0; j < 8; ++j) {
      int n = Mt * 16 + j + hi8;
      vnew[( 0 + r) * 129 + n] = a0[j];
      vnew[(16 + r) * 129 + n] = a1[j];
      vnew[(32 + r) * 129 + n] = a2[j];
      vnew[(48 + r) * 129 + n] = a3[j];
    }
    __syncthreads();

    // coalesced float4 stores with Z scaling: out[b][e][n]
    #pragma unroll
    for (int i = 0; i < 8; ++i) {
      int idx4 = t + 256 * i;
      int e  = idx4 >> 5;
      int n4 = idx4 & 31;
      float4 o;
      o.x = vnew[e * 129 + n4 * 4 + 0] * zbuf[n4 * 4 + 0];
      o.y = vnew[e * 129 + n4 * 4 + 1] * zbuf[n4 * 4 + 1];
      o.z = vnew[e * 129 + n4 * 4 + 2] * zbuf[n4 * 4 + 2];
      o.w = vnew[e * 129 + n4 * 4 + 3] * zbuf[n4 * 4 + 3];
      *(float4*)(out + gbase + (size_t)e * N_ + (size_t)c * 128 + n4 * 4) = o;
    }
    __syncthreads();   // before next chunk overwrites qt/zbuf
  }
}

extern "C" void kernel_launch(void* const* d_in, const int* in_sizes, int n_in,
                              void* d_out, int out_size, void* d_ws, size_t ws_size,
                              hipStream_t stream) {
  (void)in_sizes; (void)n_in; (void)out_size;
  const float* q      = (const float*)d_in[0];
  const float* k      = (const float*)d_in[1];
  const float* v      = (const float*)d_in[2];
  const float* scaleK = (const float*)d_in[3];
  float* out = (float*)d_out;

  // workspace: B*wgs1 partials + B reduced descriptors (deterministic sizing)
  size_t ws_floats = ws_size / sizeof(float);
  int wgs1 = 128;
  while (wgs1 > 1 && (size_t)B_ * (size_t)(wgs1 + 1) * PART_STRIDE > ws_floats)
    wgs1 >>= 1;
  float* partials = (float*)d_ws;
  float* red      = partials + (size_t)B_ * wgs1 * PART_STRIDE;

  la_pass1<<<dim3(wgs1, B_), 256, 0, stream>>>(k, v, scaleK, partials);
  la_reduce<<<(B_ * PART_STRIDE + 255) / 256, 256, 0, stream>>>(partials, red, wgs1);
  la_pass2<<<dim3(256, B_), 256, 0, stream>>>(q, red, out);
}